// GNNModel_68934225101487
// MI455X (gfx1250) — compile-verified
//
#include <hip/hip_runtime.h>
#include <hip/hip_bf16.h>
#include <stdint.h>

typedef __attribute__((ext_vector_type(2))) float v2f;
typedef __attribute__((ext_vector_type(8))) float v8f;

#define IN_DIM 29
#define HID_DIM 64
#define OUT_DIM 2

// ---------------- degree / norm ----------------

__global__ void k_deg_init(float* __restrict__ deg, int N) {
    int i = blockIdx.x * blockDim.x + threadIdx.x;
    if (i < N) deg[i] = 1.0f;  // self-loop contribution
}

__global__ void k_deg_accum(const long long* __restrict__ dst, float* __restrict__ deg, int E) {
    int e = blockIdx.x * blockDim.x + threadIdx.x;
    if (e < E) unsafeAtomicAdd(&deg[(int)dst[e]], 1.0f);
}

__global__ void k_rsqrt_inplace(float* __restrict__ deg, int N) {
    int i = blockIdx.x * blockDim.x + threadIdx.x;
    if (i < N) deg[i] = rsqrtf(deg[i]);  // deg >= 1 always
}

// ---------------- layer 1 dense transform: h = x @ W1 via V_WMMA_F32_16X16X4_F32 ----------------
// One wave -> 16 rows x 64 cols. Block = 256 threads = 8 waves = 128 rows.
// All loads unconditional (row clamped; K-tail masked by lane select) so EXEC
// stays all-ones with no saveexec blocks around the WMMA feed loads.

__global__ void k_gemm1_wmma(const float* __restrict__ x, const float* __restrict__ W,
                             float* __restrict__ h, int N) {
    const int lane = threadIdx.x & 31;
    const int wave = threadIdx.x >> 5;
    const int row0 = blockIdx.x * 128 + wave * 16;
    if (row0 >= N) return;                 // wave-uniform exit: EXEC all-ones for WMMA

    const int m  = lane & 15;
    const int hi = lane >> 4;              // 0: K=k0,k0+1  1: K=k0+2,k0+3
    const int row   = row0 + m;
    const int rload = row < N ? row : N - 1;     // clamp: A row M only feeds D row M,
    const float* xr = x + (size_t)rload * IN_DIM; // and rows >= N are dropped at store

    v8f acc[4] = {v8f{}, v8f{}, v8f{}, v8f{}};

    // kk = 0..6: every lane K-index <= 27 < 29, fully in range -> unconditional loads
#pragma unroll
    for (int kk = 0; kk < 7; ++kk) {
        const int ka = kk * 4 + hi * 2;
        v2f a;
        a.x = xr[ka];
        a.y = xr[ka + 1];
#pragma unroll
        for (int nt = 0; nt < 4; ++nt) {
            v2f b;
            b.x = W[(size_t)(ka)     * HID_DIM + nt * 16 + m];
            b.y = W[(size_t)(ka + 1) * HID_DIM + nt * 16 + m];
            acc[nt] = __builtin_amdgcn_wmma_f32_16x16x4_f32(
                false, a, false, b, (short)0, acc[nt], false, false);
        }
    }

    // kk = 7: K = 28..31; only K=28 is real. Both lane halves load the same
    // in-bounds K=28 element, hi-half zeroed by select (v_cndmask, no branch).
    {
        const float av = xr[28];
        v2f a;
        a.x = hi ? 0.0f : av;
        a.y = 0.0f;
#pragma unroll
        for (int nt = 0; nt < 4; ++nt) {
            const float bv = W[(size_t)28 * HID_DIM + nt * 16 + m];
            v2f b;
            b.x = hi ? 0.0f : bv;
            b.y = 0.0f;
            acc[nt] = __builtin_amdgcn_wmma_f32_16x16x4_f32(
                false, a, false, b, (short)0, acc[nt], false, false);
        }
    }

    // C layout: element i of v8f in lane L -> row = row0 + (L<16 ? i : i+8), col = n0 + (L&15)
    const int rbase = row0 + hi * 8;
    if (row0 + 16 <= N) {                  // wave-uniform full-tile fast path
#pragma unroll
        for (int nt = 0; nt < 4; ++nt) {
            const int col = nt * 16 + m;
#pragma unroll
            for (int i = 0; i < 8; ++i)
                h[(size_t)(rbase + i) * HID_DIM + col] = acc[nt][i];
        }
    } else {                               // tail tile: per-row guard
#pragma unroll
        for (int nt = 0; nt < 4; ++nt) {
            const int col = nt * 16 + m;
#pragma unroll
            for (int i = 0; i < 8; ++i)
                if (rbase + i < N) h[(size_t)(rbase + i) * HID_DIM + col] = acc[nt][i];
        }
    }
}

// ---------------- layer 1: bias + self-loop init, then edge scatter ----------------

__global__ void k_self1(const float* __restrict__ h, const float* __restrict__ dinv,
                        const float* __restrict__ b1, float* __restrict__ out1, int N) {
    int i = blockIdx.x * blockDim.x + threadIdx.x;   // over N*64
    if (i < N * HID_DIM) {
        int n = i >> 6;
        int c = i & 63;
        float di = dinv[n];
        out1[i] = b1[c] + h[i] * di * di;
    }
}

__global__ void k_edge1(const long long* __restrict__ ei, const float* __restrict__ dinv,
                        const float* __restrict__ h, float* __restrict__ out1, int E) {
    long long gid = (long long)blockIdx.x * blockDim.x + threadIdx.x;
    int e = (int)(gid >> 4);
    if (e >= E) return;
    int q = (int)(gid & 15);                 // 16 threads per edge, 4 cols each
    int s = (int)ei[e];
    int d = (int)ei[(size_t)E + e];
    float w = dinv[s] * dinv[d];
    const float4 hv = *(const float4*)(h + (size_t)s * HID_DIM + q * 4);
    float* o = out1 + (size_t)d * HID_DIM + q * 4;
    unsafeAtomicAdd(o + 0, hv.x * w);
    unsafeAtomicAdd(o + 1, hv.y * w);
    unsafeAtomicAdd(o + 2, hv.z * w);
    unsafeAtomicAdd(o + 3, hv.w * w);
}

// ---------------- layer 2 dense transform (ReLU fused): t2 = relu(out1) @ W2 ----------------

__global__ void k_gemm2(const float* __restrict__ out1, const float* __restrict__ W2,
                        float* __restrict__ t2, int N) {
    __shared__ float sw[HID_DIM * OUT_DIM];
    if (threadIdx.x < HID_DIM * OUT_DIM) sw[threadIdx.x] = W2[threadIdx.x];
    __syncthreads();
    int n = blockIdx.x * blockDim.x + threadIdx.x;
    if (n >= N) return;
    const float* row = out1 + (size_t)n * HID_DIM;
    float a0 = 0.0f, a1 = 0.0f;
#pragma unroll 8
    for (int k = 0; k < HID_DIM; ++k) {
        float v = row[k];
        v = v > 0.0f ? v : 0.0f;             // ReLU
        a0 = fmaf(v, sw[2 * k + 0], a0);
        a1 = fmaf(v, sw[2 * k + 1], a1);
    }
    t2[2 * (size_t)n + 0] = a0;
    t2[2 * (size_t)n + 1] = a1;
}

// ---------------- layer 2: bias + self-loop init, then edge scatter ----------------

__global__ void k_self2(const float* __restrict__ t2, const float* __restrict__ dinv,
                        const float* __restrict__ b2, float* __restrict__ out, int N) {
    int n = blockIdx.x * blockDim.x + threadIdx.x;
    if (n >= N) return;
    float di = dinv[n];
    float ds = di * di;
    out[2 * (size_t)n + 0] = b2[0] + t2[2 * (size_t)n + 0] * ds;
    out[2 * (size_t)n + 1] = b2[1] + t2[2 * (size_t)n + 1] * ds;
}

__global__ void k_edge2(const long long* __restrict__ ei, const float* __restrict__ dinv,
                        const float* __restrict__ t2, float* __restrict__ out, int E) {
    int e = blockIdx.x * blockDim.x + threadIdx.x;
    if (e >= E) return;
    int s = (int)ei[e];
    int d = (int)ei[(size_t)E + e];
    float w = dinv[s] * dinv[d];
    float2 tv = *(const float2*)(t2 + 2 * (size_t)s);
    unsafeAtomicAdd(&out[2 * (size_t)d + 0], tv.x * w);
    unsafeAtomicAdd(&out[2 * (size_t)d + 1], tv.y * w);
}

// ---------------- host launch ----------------

extern "C" void kernel_launch(void* const* d_in, const int* in_sizes, int n_in,
                              void* d_out, int out_size, void* d_ws, size_t ws_size,
                              hipStream_t stream) {
    const float*     x  = (const float*)d_in[0];       // [N, 29]
    const long long* ei = (const long long*)d_in[1];   // [2, E] int64
    const float*     W1 = (const float*)d_in[2];       // [29, 64]
    const float*     b1 = (const float*)d_in[3];       // [64]
    const float*     W2 = (const float*)d_in[4];       // [64, 2]
    const float*     b2 = (const float*)d_in[5];       // [2]
    float*           out = (float*)d_out;              // [N, 2]

    const int N = in_sizes[0] / IN_DIM;
    const int E = in_sizes[1] / 2;

    // workspace carve-out (256B aligned): dinv N, h N*64, out1 N*64, t2 N*2  (~52.4 MB)
    size_t off = 0;
    char* base = (char*)d_ws;
    auto carve = [&](size_t bytes) -> float* {
        float* p = (float*)(base + off);
        off += (bytes + 255) & ~(size_t)255;
        return p;
    };
    float* dinv = carve((size_t)N * 4);
    float* h    = carve((size_t)N * HID_DIM * 4);
    float* out1 = carve((size_t)N * HID_DIM * 4);
    float* t2   = carve((size_t)N * OUT_DIM * 4);
    (void)ws_size;

    const int B = 256;
    // norm
    k_deg_init<<<(N + B - 1) / B, B, 0, stream>>>(dinv, N);
    k_deg_accum<<<(E + B - 1) / B, B, 0, stream>>>(ei + E, dinv, E);
    k_rsqrt_inplace<<<(N + B - 1) / B, B, 0, stream>>>(dinv, N);
    // layer 1
    k_gemm1_wmma<<<(N + 127) / 128, B, 0, stream>>>(x, W1, h, N);
    k_self1<<<((size_t)N * HID_DIM + B - 1) / B, B, 0, stream>>>(h, dinv, b1, out1, N);
    {
        long long t = (long long)E * 16;
        k_edge1<<<(unsigned)((t + B - 1) / B), B, 0, stream>>>(ei, dinv, h, out1, E);
    }
    // layer 2
    k_gemm2<<<(N + B - 1) / B, B, 0, stream>>>(out1, W2, t2, N);
    k_self2<<<(N + B - 1) / B, B, 0, stream>>>(t2, dinv, b2, out, N);
    k_edge2<<<(E + B - 1) / B, B, 0, stream>>>(ei, dinv, t2, out, E);
}